// GraphSAGE_33122787787018
// MI455X (gfx1250) — compile-verified
//
#include <hip/hip_runtime.h>
#include <math.h>

// ---------------------------------------------------------------------------
// GraphSAGE on gfx1250 (MI455X).
//   Layer trick: mean(x[src]) @ W == segment_sum((x@W)[src]) / cnt
//   -> project to 32 dims FIRST, then scatter 32-dim messages (4x less traffic;
//      the 12.8MB node tables are resident in the 192MB L2).
//   GEMMs: V_WMMA_F32_16X16X4_F32 (exact fp32), fully unrolled via templates,
//   dual-accumulator kernel reads A once for both W_l and W_r projections.
// ---------------------------------------------------------------------------

typedef float v2f __attribute__((ext_vector_type(2)));
typedef float v8f __attribute__((ext_vector_type(8)));

#define N_NODES 100000
#define N_EDGES 3200000

__device__ __forceinline__ v8f wmma_f32(v2f a, v2f b, v8f c) {
  return __builtin_amdgcn_wmma_f32_16x16x4_f32(
      /*neg_a=*/false, a, /*neg_b=*/false, b,
      /*c_mod=*/(short)0, c, /*reuse_a=*/false, /*reuse_b=*/false);
}

// ---- Dual GEMM: C1 = A@B1, C2 = A@B2. A:[Mx K], B:[K x 32], C:[M x 32]. ----
// One wave32 per 16x16 output tile pair; A tile loaded once, 2 WMMAs per step.
template<int K>
__global__ __launch_bounds__(32) void gemm_dual_wmma(
    const float* __restrict__ A, const float* __restrict__ B1,
    const float* __restrict__ B2, float* __restrict__ C1,
    float* __restrict__ C2)
{
  const int m0    = blockIdx.x * 16;
  const int n0    = blockIdx.y * 16;
  const int lane  = threadIdx.x;
  const int row   = lane & 15;          // A: M index
  const int khalf = (lane >> 4) << 1;   // K sub-offset 0 or 2
  const int col   = lane & 15;          // B/C: N index

  const float* ap  = A  + (size_t)(m0 + row) * K + khalf;  // a.x=K+0, a.y=K+1
  const float* b1p = B1 + (size_t)khalf * 32 + n0 + col;   // rows khalf, khalf+1
  const float* b2p = B2 + (size_t)khalf * 32 + n0 + col;

  v8f c1 = {}, c2 = {};
  #pragma unroll
  for (int k0 = 0; k0 < K; k0 += 4) {
    v2f a;  a.x  = ap[k0];            a.y  = ap[k0 + 1];
    v2f b1; b1.x = b1p[k0 * 32];      b1.y = b1p[(k0 + 1) * 32];
    v2f b2; b2.x = b2p[k0 * 32];      b2.y = b2p[(k0 + 1) * 32];
    c1 = wmma_f32(a, b1, c1);
    c2 = wmma_f32(a, b2, c2);
  }

  const int mbase = m0 + ((lane >> 4) << 3);  // D: M = v + 8*(lane>=16)
  #pragma unroll
  for (int v = 0; v < 8; ++v) {
    C1[(size_t)(mbase + v) * 32 + n0 + col] = c1[v];
    C2[(size_t)(mbase + v) * 32 + n0 + col] = c2[v];
  }
}

// ---- Single GEMM with bias + column masking (for the NCOLS=20 classifier) ----
template<int K, int NCOLS>
__global__ __launch_bounds__(32) void gemm_wmma(
    const float* __restrict__ A, const float* __restrict__ B,
    float* __restrict__ C, const float* __restrict__ bias)
{
  constexpr bool FULL = (NCOLS % 16) == 0;
  const int m0    = blockIdx.x * 16;
  const int n0    = blockIdx.y * 16;
  const int lane  = threadIdx.x;
  const int row   = lane & 15;
  const int khalf = (lane >> 4) << 1;
  const int col   = lane & 15;
  const bool colOK = FULL || (n0 + col) < NCOLS;
  const float msk  = colOK ? 1.0f : 0.0f;     // branch-free column mask
  const int   bcol = colOK ? (n0 + col) : 0;  // clamped (in-bounds) address

  const float* ap = A + (size_t)(m0 + row) * K + khalf;
  const float* bp = B + (size_t)khalf * NCOLS + bcol;

  v8f c = {};
  #pragma unroll
  for (int k0 = 0; k0 < K; k0 += 4) {
    v2f a; a.x = ap[k0]; a.y = ap[k0 + 1];
    v2f b;
    if (FULL) { b.x = bp[k0 * NCOLS];       b.y = bp[(k0 + 1) * NCOLS]; }
    else      { b.x = bp[k0 * NCOLS] * msk; b.y = bp[(k0 + 1) * NCOLS] * msk; }
    c = wmma_f32(a, b, c);
  }

  if (colOK) {
    const float bv = bias ? bias[n0 + col] : 0.0f;
    const int mbase = m0 + ((lane >> 4) << 3);
    #pragma unroll
    for (int v = 0; v < 8; ++v)
      C[(size_t)(mbase + v) * NCOLS + n0 + col] = c[v] + bv;
  }
}

__global__ __launch_bounds__(256) void zero_f32(float* __restrict__ p, size_t n) {
  size_t i = (size_t)blockIdx.x * 256 + threadIdx.x;
  if (i < n) p[i] = 0.0f;
}

// 8 threads per edge, 4 features each (float4 gather + 4 f32 global atomics).
__global__ __launch_bounds__(256) void scatter_edges(
    const float* __restrict__ p, const int* __restrict__ src,
    const int* __restrict__ dst, float* __restrict__ agg,
    float* __restrict__ cnt, int nEdges, int doCnt)
{
  int t = blockIdx.x * 256 + threadIdx.x;
  int e = t >> 3;
  if (e >= nEdges) return;
  int part = t & 7;
  int s = src[e], d = dst[e];
  const float4 v = *(const float4*)(p + (size_t)s * 32 + part * 4);
  float* ap = agg + (size_t)d * 32 + part * 4;
  unsafeAtomicAdd(ap + 0, v.x);
  unsafeAtomicAdd(ap + 1, v.y);
  unsafeAtomicAdd(ap + 2, v.z);
  unsafeAtomicAdd(ap + 3, v.w);
  if (doCnt && part == 0) unsafeAtomicAdd(cnt + d, 1.0f);
}

// One wave32 per node: v = agg/max(cnt,1) + b + r ; L2-normalize over 32 dims ; tanh.
__global__ __launch_bounds__(256) void combine_norm_tanh(
    const float* __restrict__ agg, const float* __restrict__ cnt,
    const float* __restrict__ bias, const float* __restrict__ r,
    float* __restrict__ h, int n)
{
  int node = blockIdx.x * 8 + (threadIdx.x >> 5);
  int lane = threadIdx.x & 31;
  if (node >= n) return;
  float cdeg = fmaxf(cnt[node], 1.0f);
  float v = agg[(size_t)node * 32 + lane] / cdeg + bias[lane]
          + r[(size_t)node * 32 + lane];
  float s = v * v;
  #pragma unroll
  for (int off = 16; off >= 1; off >>= 1) s += __shfl_xor(s, off, 32);
  float nrm = fmaxf(sqrtf(s), 1e-12f);
  h[(size_t)node * 32 + lane] = tanhf(v / nrm);
}

// One thread per node over 20 logits (already biased by the GEMM).
__global__ __launch_bounds__(256) void log_softmax20(
    const float* __restrict__ logits, float* __restrict__ out, int n)
{
  int i = blockIdx.x * 256 + threadIdx.x;
  if (i >= n) return;
  float v[20];
  float m = -INFINITY;
  #pragma unroll
  for (int j = 0; j < 20; ++j) { v[j] = logits[(size_t)i * 20 + j]; m = fmaxf(m, v[j]); }
  float s = 0.0f;
  #pragma unroll
  for (int j = 0; j < 20; ++j) s += expf(v[j] - m);
  float lse = m + logf(s);
  #pragma unroll
  for (int j = 0; j < 20; ++j) out[(size_t)i * 20 + j] = v[j] - lse;
}

extern "C" void kernel_launch(void* const* d_in, const int* in_sizes, int n_in,
                              void* d_out, int out_size, void* d_ws, size_t ws_size,
                              hipStream_t stream) {
  const float* x   = (const float*)d_in[0];
  const int*   ei  = (const int*)  d_in[1];
  const float* W1l = (const float*)d_in[2];
  const float* b1  = (const float*)d_in[3];
  const float* W1r = (const float*)d_in[4];
  const float* W2l = (const float*)d_in[5];
  const float* b2  = (const float*)d_in[6];
  const float* W2r = (const float*)d_in[7];
  const float* Wc  = (const float*)d_in[8];
  const float* bc  = (const float*)d_in[9];
  float* out = (float*)d_out;

  const int N = N_NODES, E = N_EDGES;
  const int* src = ei;
  const int* dst = ei + E;

  // Workspace layout (floats): P | R | AGG | CNT | H1    (~51.6 MB total)
  float* ws  = (float*)d_ws;
  float* P   = ws;
  float* R   = ws + (size_t)N * 32;
  float* AGG = ws + (size_t)2 * N * 32;
  float* CNT = ws + (size_t)3 * N * 32;      // contiguous after AGG
  float* H1  = ws + (size_t)3 * N * 32 + N;
  float* LOGITS = P;                          // reuse P after layer 2
  float* H2  = out + (size_t)N * 20;          // second tuple output lives in d_out

  dim3 gTile(N / 16, 2);        // N%16 == 0; 32 (or 20-masked) cols = 2 N-tiles
  int zBlocks1 = (int)(((size_t)N * 32 + N + 255) / 256);
  int zBlocks2 = (int)(((size_t)N * 32 + 255) / 256);
  int sBlocks  = (int)(((size_t)E * 8 + 255) / 256);

  // ---- Layer 1 ----
  zero_f32<<<zBlocks1, 256, 0, stream>>>(AGG, (size_t)N * 32 + N);  // AGG + CNT
  gemm_dual_wmma<128><<<gTile, 32, 0, stream>>>(x, W1l, W1r, P, R);
  scatter_edges<<<sBlocks, 256, 0, stream>>>(P, src, dst, AGG, CNT, E, 1);
  combine_norm_tanh<<<N / 8, 256, 0, stream>>>(AGG, CNT, b1, R, H1, N);

  // ---- Layer 2 (cnt unchanged: same edges) ----
  zero_f32<<<zBlocks2, 256, 0, stream>>>(AGG, (size_t)N * 32);
  gemm_dual_wmma<32><<<gTile, 32, 0, stream>>>(H1, W2l, W2r, P, R);
  scatter_edges<<<sBlocks, 256, 0, stream>>>(P, src, dst, AGG, CNT, E, 0);
  combine_norm_tanh<<<N / 8, 256, 0, stream>>>(AGG, CNT, b2, R, H2, N);

  // ---- Classifier + log_softmax ----
  gemm_wmma<32, 20><<<gTile, 32, 0, stream>>>(H2, Wc, LOGITS, bc);
  log_softmax20<<<(N + 255) / 256, 256, 0, stream>>>(LOGITS, out, N);
}